// NcutLoss_old_75952201663247
// MI455X (gfx1250) — compile-verified
//
#include <hip/hip_runtime.h>

// ---------------- problem constants (from reference) ----------------
#define RR       5
#define NN       4
#define KK       8
#define CC       3
#define HH       1024
#define WW_      1024
#define SS       1000
#define NP       81            // lattice points with i^2+j^2 <= 25
#define HW       (HH * WW_)

// 256 blocks x 8 wave32 = 2048 waves; ~2 samples per wave.  Gather-latency
// bound kernel -> maximize resident waves per SIMD32.
#define BLOCKS_PER_N     64
#define WAVES_PER_BLOCK  8
#define WAVES_PER_N      (BLOCKS_PER_N * WAVES_PER_BLOCK)   // 512

// ---------------- compile-time disk offset table ----------------
struct Disk { signed char di[NP]; signed char dj[NP]; };
constexpr Disk makeDisk() {
    Disk d{};
    int n = 0;
    for (int i = -RR; i <= RR; ++i)
        for (int j = -RR; j <= RR; ++j)
            if (i * i + j * j <= RR * RR) {
                d.di[n] = (signed char)i;
                d.dj[n] = (signed char)j;
                ++n;
            }
    return d;
}
__constant__ Disk kDisk = makeDisk();

typedef __attribute__((ext_vector_type(2))) float v2f;
typedef __attribute__((ext_vector_type(8))) float v8f;

__device__ __forceinline__ float waveSum32(float v) {
#pragma unroll
    for (int m = 16; m > 0; m >>= 1) v += __shfl_xor(v, m, 32);
    return v;
}

// ---------------- zero the 64-float accumulator in d_ws ----------------
__global__ void ncut_zero(float* __restrict__ acc) {
    if (threadIdx.x < NN * 16) acc[threadIdx.x] = 0.f;
}

// ---------------- main gather + accumulate kernel ----------------
__global__ __launch_bounds__(256) void ncut_main(
    const float* __restrict__ pred,      // (N,K,H,W)
    const float* __restrict__ imgs,      // (N,C,H,W)
    const long long* __restrict__ hs,    // (S,) int64
    const long long* __restrict__ wsmp,  // (S,) int64
    float* __restrict__ acc)             // (N,16): [num0..7, den0..7]
{
    __shared__ float lds[WAVES_PER_BLOCK * 16];

    const int n       = blockIdx.x / BLOCKS_PER_N;
    const int blkInN  = blockIdx.x % BLOCKS_PER_N;
    const int wid     = threadIdx.x >> 5;
    const int lane    = threadIdx.x & 31;
    const int waveInN = blkInN * WAVES_PER_BLOCK + wid;

    const float* __restrict__ imgN  = imgs + (size_t)n * CC * HW;
    const float* __restrict__ predN = pred + (size_t)n * KK * HW;

    // ---- sample-invariant per-lane disk geometry (hoisted out of s-loop) ----
    int   dOff[3];   // di*W + dj
    float rTerm[3];  // r^2 / sigma_x^2 (already scaled), inf-free
    bool  act[3];
#pragma unroll
    for (int pass = 0; pass < 3; ++pass) {
        const int p  = pass * 32 + lane;
        act[pass]    = (p < NP);
        const int pi = act[pass] ? p : 0;
        const int di = kDisk.di[pi];
        const int dj = kDisk.dj[pi];
        dOff[pass]   = di * WW_ + dj;
        rTerm[pass]  = (float)(di * di + dj * dj) * 0.0625f;
    }

    float accNum[KK];
    float accDen[KK];
#pragma unroll
    for (int k = 0; k < KK; ++k) { accNum[k] = 0.f; accDen[k] = 0.f; }

    for (int s = waveInN; s < SS; s += WAVES_PER_N) {
        const int h    = (int)hs[s] + RR;
        const int w    = (int)wsmp[s] + RR;
        const int cOff = h * WW_ + w;

        // prefetch the next sample's patch rows (gfx1250 global_prefetch_b8)
        const int sn = s + WAVES_PER_N;
        if (sn < SS) {
            const int hn   = (int)hs[sn] + RR;
            const int wn   = (int)wsmp[sn] + RR;
            const int offn = hn * WW_ + (wn - RR);
            __builtin_prefetch(imgN + offn, 0, 3);
            __builtin_prefetch(imgN + HW + offn, 0, 3);
            __builtin_prefetch(imgN + 2 * HW + offn, 0, 3);
            __builtin_prefetch(predN + offn, 0, 3);
        }

        // center values (uniform address -> broadcast request)
        const float c0 = imgN[cOff];
        const float c1 = imgN[HW + cOff];
        const float c2 = imgN[2 * HW + cOff];
        float pc[KK];
#pragma unroll
        for (int k = 0; k < KK; ++k) pc[k] = predN[k * HW + cOff];

        float laneW = 0.f;
        float laneT[KK];
#pragma unroll
        for (int k = 0; k < KK; ++k) laneT[k] = 0.f;

#pragma unroll
        for (int pass = 0; pass < 3; ++pass) {
            const int off = cOff + dOff[pass];

            const float a0 = imgN[off] - c0;
            const float a1 = imgN[HW + off] - c1;
            const float a2 = imgN[2 * HW + off] - c2;
            const float d2 = a0 * a0 + a1 * a1 + a2 * a2;
            // exp(-d2/sigmaI^2)*exp(-r2/sigmaX^2) fused into one v_exp_f32
            float wgt = __expf(-(d2 * 0.01f + rTerm[pass]));
            if (!act[pass]) wgt = 0.f;

            laneW += wgt;
#pragma unroll
            for (int k = 0; k < KK; ++k)
                laneT[k] = fmaf(wgt, predN[k * HW + off], laneT[k]);
        }

#pragma unroll
        for (int k = 0; k < KK; ++k) {
            accNum[k] = fmaf(pc[k], laneT[k], accNum[k]);
            accDen[k] = fmaf(pc[k], laneW, accDen[k]);
        }
    }

    // ---- per-wave cross-lane butterfly: 16 values over 32 lanes ----
    float red[16];
#pragma unroll
    for (int j = 0; j < 16; ++j) {
        const float v = (j < 8) ? accNum[j] : accDen[j - 8];
        red[j] = waveSum32(v);
    }
    if (lane < 16) {
        float mine = 0.f;
#pragma unroll
        for (int j = 0; j < 16; ++j) mine = (lane == j) ? red[j] : mine;
        lds[wid * 16 + lane] = mine;
    }
    __syncthreads();

    // ---- cross-wave reduction on the matrix pipe ----
    // D(16x16) += A(16x4) x B(4x16);  A[m,kidx] = wave kidx's partial m,
    // B = all-ones -> column 0 of D = sum over the block's 8 waves.
    // EXEC is all-ones for wave 0 (threads 0..31).
    if (threadIdx.x < 32) {
        const int l  = threadIdx.x;
        const int m  = l & 15;   // A-matrix row (M)
        const int hi = l >> 4;   // 0 -> K=0,1 ; 1 -> K=2,3

        v2f B;  B[0] = 1.f; B[1] = 1.f;
        v2f A0, A1;
        A0[0] = lds[(hi * 2 + 0) * 16 + m];
        A0[1] = lds[(hi * 2 + 1) * 16 + m];
        A1[0] = lds[(4 + hi * 2 + 0) * 16 + m];
        A1[1] = lds[(4 + hi * 2 + 1) * 16 + m];

        v8f c = {};
        c = __builtin_amdgcn_wmma_f32_16x16x4_f32(false, A0, false, B,
                                                  (short)0, c, false, false);
        c = __builtin_amdgcn_wmma_f32_16x16x4_f32(false, A1, false, B,
                                                  (short)0, c, false, false);

        // D[M=r+8*hi, N=0] lives in VGPR r of lanes 0 (M=0..7) and 16 (M=8..15)
        if (m == 0) {
            float* base = acc + n * 16 + hi * 8;
#pragma unroll
            for (int r = 0; r < 8; ++r) atomicAdd(base + r, c[r]);
        }
    }
}

// ---------------- finalize: loss = N*K - sum(num/den) ----------------
__global__ void ncut_final(const float* __restrict__ acc, float* __restrict__ out) {
    const int l = threadIdx.x;
    float v = 0.f;
    if (l < NN * KK) {
        const int n = l >> 3, k = l & 7;
        v = acc[n * 16 + k] / acc[n * 16 + 8 + k];
    }
    v = waveSum32(v);
    if (l == 0) out[0] = (float)(NN * KK) - v;
}

extern "C" void kernel_launch(void* const* d_in, const int* in_sizes, int n_in,
                              void* d_out, int out_size, void* d_ws, size_t ws_size,
                              hipStream_t stream) {
    const float*     pred = (const float*)d_in[0];
    const float*     imgs = (const float*)d_in[1];
    const long long* hs   = (const long long*)d_in[2];
    const long long* wsmp = (const long long*)d_in[3];
    float* acc = (float*)d_ws;   // 64 floats
    float* out = (float*)d_out;  // 1 float

    ncut_zero<<<1, 64, 0, stream>>>(acc);
    ncut_main<<<dim3(NN * BLOCKS_PER_N), 256, 0, stream>>>(pred, imgs, hs, wsmp, acc);
    ncut_final<<<1, 32, 0, stream>>>(acc, out);
}